// E2EGuidedFilter_72206990181004
// MI455X (gfx1250) — compile-verified
//
#include <hip/hip_runtime.h>

// Fused guided filter for MI455X (gfx1250, wave32).
// One kernel, one read of x/y, one write of out (~302 MB -> ~13us HBM floor).
// All 6 box filters run as banded-matrix GEMMs on the WMMA pipe
// (V_WMMA_F32_16X16X4_F32); elementwise math on VALU.
// Round-3: bulk x/y staging now uses the CDNA5 async copy path
// (GLOBAL_LOAD_ASYNC_TO_LDS_B128 + s_wait_asynccnt) instead of a
// global->VGPR->LDS round trip; zero-padded border slots keep the VGPR path.

typedef __attribute__((ext_vector_type(2))) float v2f;
typedef __attribute__((ext_vector_type(8))) float v8f;

constexpr int   GF_H   = 1024;
constexpr int   GF_W   = 1024;
constexpr int   RAD    = 8;
constexpr float GF_EPS = 0.01f;

constexpr int TS   = 16;          // WMMA tile side
constexpr int NT   = 4;           // tiles per region side
constexpr int REG  = NT * TS;     // 64: halo region side
constexpr int OUTT = 32;          // output tile side
constexpr int LDSW = REG + 1;     // 65: padded LDS row stride (odd -> bank friendly)
constexpr int ASZ  = REG * LDSW;  // floats per LDS array (4160)

__device__ __forceinline__ v8f wmma4(v2f a, v2f b, v8f c) {
    // (neg_a, A, neg_b, B, c_mod, C, reuse_a, reuse_b)
    return __builtin_amdgcn_wmma_f32_16x16x4_f32(false, a, false, b, (short)0, c,
                                                 false, false);
}

// Async DMA of 16 bytes global -> LDS (ASYNCcnt-tracked, no VGPR round trip).
// lds_byte_off is the LDS byte address (dynamic LDS block starts at offset 0).
__device__ __forceinline__ void async_ld_b128(unsigned lds_byte_off, const float* g) {
    asm volatile("global_load_async_to_lds_b128 %0, %1, off"
                 :: "v"(lds_byte_off), "v"(g)
                 : "memory");
}
__device__ __forceinline__ void wait_async0() {
    asm volatile("s_wait_asynccnt 0" ::: "memory");
}

// Banded box-filter weight: 1 if source index (k + 16*nb) is within RAD of dest n.
__device__ __forceinline__ float bandw(int k, int n, int nb) {
    const int d = k + TS * nb - n;
    return (d >= -RAD && d <= RAD) ? 1.0f : 0.0f;
}

// C/D layout: VGPR v, lanes 0-15 -> row v, col lane; lanes 16-31 -> row v+8.
__device__ __forceinline__ void store_tile(float* dst, int tr, int tc, int m,
                                           int half, v8f acc) {
#pragma unroll
    for (int v = 0; v < 8; ++v)
        dst[(tr * TS + v + 8 * half) * LDSW + tc * TS + m] = acc[v];
}

// Horizontal box-sum, quantity chosen at compile time:
//   Q=0: x    Q=1: y    Q=2: x*y    Q=3: x*x    Q=4: raw src (second stage)
// D = sum_nb T(tc+nb) * W(nb).  A = data (lane m = row M; VGPR v holds K=k0+v),
// B = constant band fragments from the hoisted register table bw.
template <int Q>
__device__ __forceinline__ void hbox(const float* Xs, const float* Ys, float* dst,
                                     int tr, int tc, int m, int half,
                                     const v2f (&bw)[3][4]) {
    v8f acc = {0.f, 0.f, 0.f, 0.f, 0.f, 0.f, 0.f, 0.f};
    const int row = tr * TS + m;
#pragma unroll
    for (int nb = -1; nb <= 1; ++nb) {
        const int tcn = tc + nb;                   // SGPR (tr/tc uniform)
        if (tcn < 0 || tcn >= NT) continue;        // scalar branch, EXEC untouched
        const int cb = tcn * TS;
#pragma unroll
        for (int kk = 0; kk < 4; ++kk) {
            const int k0  = kk * 4 + 2 * half;
            const int idx = row * LDSW + cb + k0;
            v2f A;
            if constexpr (Q == 0)      { A.x = Xs[idx];           A.y = Xs[idx + 1]; }
            else if constexpr (Q == 1) { A.x = Ys[idx];           A.y = Ys[idx + 1]; }
            else if constexpr (Q == 2) { A.x = Xs[idx] * Ys[idx];
                                         A.y = Xs[idx + 1] * Ys[idx + 1]; }
            else if constexpr (Q == 3) { A.x = Xs[idx] * Xs[idx];
                                         A.y = Xs[idx + 1] * Xs[idx + 1]; }
            else                       { A.x = Xs[idx];           A.y = Xs[idx + 1]; }
            acc = wmma4(A, bw[nb + 1][kk], acc);
        }
    }
    store_tile(dst, tr, tc, m, half, acc);
}

// Vertical box-sum: D = sum_nb W(nb) * H(tr+nb).  A = constant band (same table),
// B = data (lane m = col N; VGPR v holds source row k0+v).
__device__ __forceinline__ void vbox(const float* src, float* dst,
                                     int tr, int tc, int m, int half,
                                     const v2f (&bw)[3][4]) {
    v8f acc = {0.f, 0.f, 0.f, 0.f, 0.f, 0.f, 0.f, 0.f};
    const int cb = tc * TS + m;
#pragma unroll
    for (int nb = -1; nb <= 1; ++nb) {
        const int trn = tr + nb;
        if (trn < 0 || trn >= NT) continue;
        const int rb = trn * TS;
#pragma unroll
        for (int kk = 0; kk < 4; ++kk) {
            const int k0 = kk * 4 + 2 * half;
            v2f B; B.x = src[(rb + k0) * LDSW + cb];
                   B.y = src[(rb + k0 + 1) * LDSW + cb];
            acc = wmma4(bw[nb + 1][kk], B, acc);
        }
    }
    store_tile(dst, tr, tc, m, half, acc);
}

__device__ __forceinline__ float cnt1(int i, int n) {
    int lo = i - RAD; if (lo < 0) lo = 0;
    int hi = i + RAD; if (hi > n - 1) hi = n - 1;
    return (float)(hi - lo + 1);
}

__global__ __launch_bounds__(256)
void E2EGuidedFilter_gfx1250_kernel(const float* __restrict__ xin,
                                    const float* __restrict__ yin,
                                    float* __restrict__ outp) {
    extern __shared__ float lds[];
    float* Xs  = lds;             // region x            (kept for final combine)
    float* Ys  = lds + 1 * ASZ;   // region y
    float* Ht  = lds + 2 * ASZ;   // horizontal-sum temp
    float* Sx  = lds + 3 * ASZ;   // box(x)  -> later a
    float* Sy  = lds + 4 * ASZ;   // box(y)  -> later b
    float* Sxy = lds + 5 * ASZ;   // box(xy) -> later box(a)
    float* Sxx = lds + 6 * ASZ;   // box(xx) -> later box(b)

    const int tid  = threadIdx.x;
    const int lane = tid & 31;
    // Launder the wave id into an SGPR so all tile-selection control flow is
    // provably uniform -> scalar branches only around WMMA.
    const int wave = __builtin_amdgcn_readfirstlane(tid >> 5);   // 0..7
    const int m    = lane & 15;
    const int half = lane >> 4;

    // Constant band-matrix fragments, hoisted once into registers.
    v2f bw[3][4];
#pragma unroll
    for (int nb = 0; nb < 3; ++nb)
#pragma unroll
        for (int kk = 0; kk < 4; ++kk) {
            const int k0 = kk * 4 + 2 * half;
            bw[nb][kk].x = bandw(k0,     m, nb - 1);
            bw[nb][kk].y = bandw(k0 + 1, m, nb - 1);
        }

    const int gi0 = (int)blockIdx.y * OUTT - 2 * RAD;   // region origin (may be <0)
    const int gj0 = (int)blockIdx.x * OUTT - 2 * RAD;
    const size_t pbase = (size_t)blockIdx.z * (size_t)GF_H * (size_t)GF_W;

    // ---- Stage A: stage the 64x64 x/y region into LDS.
    // Interior float4 slots: GLOBAL_LOAD_ASYNC_TO_LDS_B128 (ASYNCcnt path).
    // Border slots (zero padding): VGPR fallback + ds_store.
    for (int s = tid; s < REG * (REG / 4); s += 256) {
        const int r  = s >> 4;
        const int c  = (s & 15) << 2;
        const int gi = gi0 + r;
        const int gj = gj0 + c;
        if (gi >= 0 && gi < GF_H && gj >= 0 && gj + 3 < GF_W) {
            const size_t o = pbase + (size_t)gi * GF_W + gj;   // 16B aligned
            const unsigned loff = (unsigned)(r * LDSW + c) * 4u;
            async_ld_b128(0u * ASZ * 4u + loff, xin + o);      // -> Xs
            async_ld_b128(1u * ASZ * 4u + loff, yin + o);      // -> Ys
        } else {
            float4 xv = {0.f, 0.f, 0.f, 0.f};
            float4 yv = {0.f, 0.f, 0.f, 0.f};
            if (gi >= 0 && gi < GF_H) {
                float* xe = &xv.x; float* ye = &yv.x;
                for (int e = 0; e < 4; ++e) {
                    const int gje = gj + e;
                    if (gje >= 0 && gje < GF_W) {
                        const size_t o = pbase + (size_t)gi * GF_W + gje;
                        xe[e] = xin[o]; ye[e] = yin[o];
                    }
                }
            }
            const int p = r * LDSW + c;
            Xs[p] = xv.x; Xs[p + 1] = xv.y; Xs[p + 2] = xv.z; Xs[p + 3] = xv.w;
            Ys[p] = yv.x; Ys[p + 1] = yv.y; Ys[p + 2] = yv.z; Ys[p + 3] = yv.w;
        }
    }
    wait_async0();          // wave's async copies landed in LDS
    __syncthreads();        // + ds stores from all waves visible

    // ---- Stage B/C: separable WMMA box sums of x, y, x*y, x*x.
    // Each wave owns tiles {wave, wave+8} of the 4x4 grid; fully unrolled.
    const int t0r = wave >> 2, t0c = wave & 3;          // tile 'wave'
    const int t1r = t0r + 2,   t1c = t0c;               // tile 'wave + 8'

#define GF_STAGE_BC(Q, DST)                                                  \
    do {                                                                     \
        hbox<Q>(Xs, Ys, Ht, t0r, t0c, m, half, bw);                          \
        hbox<Q>(Xs, Ys, Ht, t1r, t1c, m, half, bw);                          \
        __syncthreads();                                                     \
        vbox(Ht, DST, t0r, t0c, m, half, bw);                                \
        vbox(Ht, DST, t1r, t1c, m, half, bw);                                \
        __syncthreads();                                                     \
    } while (0)

    GF_STAGE_BC(0, Sx);
    GF_STAGE_BC(1, Sy);
    GF_STAGE_BC(2, Sxy);
    GF_STAGE_BC(3, Sxx);
#undef GF_STAGE_BC

    // ---- Per-pixel a, b on the valid band [8,56) (zero elsewhere; also zero
    //      outside the image so stage-2 zero-padded sums match the clamped ref).
    for (int s = tid; s < REG * REG; s += 256) {
        const int r  = s >> 6;
        const int c  = s & 63;
        const int gi = gi0 + r, gj = gj0 + c;
        const int p  = r * LDSW + c;
        float av = 0.f, bv = 0.f;
        if (r >= RAD && r < REG - RAD && c >= RAD && c < REG - RAD &&
            gi >= 0 && gi < GF_H && gj >= 0 && gj < GF_W) {
            const float inv = 1.0f / (cnt1(gi, GF_H) * cnt1(gj, GF_W));
            const float xm  = Sx[p]  * inv;
            const float ym  = Sy[p]  * inv;
            const float cov = Sxy[p] * inv - xm * ym;
            const float var = Sxx[p] * inv - xm * xm;
            av = cov / (var + GF_EPS);
            bv = ym - av * xm;
        }
        Sx[p] = av;     // a  (S-buffers are dead, reuse in place)
        Sy[p] = bv;     // b
    }
    __syncthreads();

    // ---- Stage D/E: WMMA box sums of a then b (center output tiles only).
    const int dr = wave >> 1, dc = 1 + (wave & 1);      // 8 tiles, one per wave
    hbox<4>(Sx, Sx, Ht, dr, dc, m, half, bw);           // a
    __syncthreads();
    if (wave < 4)                                       // scalar (uniform) branch
        vbox(Ht, Sxy, 1 + (wave >> 1), 1 + (wave & 1), m, half, bw);
    __syncthreads();
    hbox<4>(Sy, Sy, Ht, dr, dc, m, half, bw);           // b
    __syncthreads();
    if (wave < 4)
        vbox(Ht, Sxx, 1 + (wave >> 1), 1 + (wave & 1), m, half, bw);
    __syncthreads();

    // ---- Final combine: out = mean(a)*x + mean(b) on the 32x32 center
    for (int s = tid; s < OUTT * OUTT; s += 256) {
        const int r  = 2 * RAD + (s >> 5);
        const int c  = 2 * RAD + (s & 31);
        const int gi = gi0 + r, gj = gj0 + c;    // always in-bounds (1024 % 32 == 0)
        const int p  = r * LDSW + c;
        const float inv = 1.0f / (cnt1(gi, GF_H) * cnt1(gj, GF_W));
        outp[pbase + (size_t)gi * GF_W + gj] = (Sxy[p] * inv) * Xs[p] + Sxx[p] * inv;
    }
}

extern "C" void kernel_launch(void* const* d_in, const int* in_sizes, int n_in,
                              void* d_out, int out_size, void* d_ws, size_t ws_size,
                              hipStream_t stream) {
    (void)n_in; (void)out_size; (void)d_ws; (void)ws_size;
    const float* x = (const float*)d_in[0];
    const float* y = (const float*)d_in[1];
    float* o = (float*)d_out;

    const int planes = in_sizes[0] / (GF_H * GF_W);          // 8*3 = 24
    dim3 grid(GF_W / OUTT, GF_H / OUTT, planes);             // 32 x 32 x 24
    const size_t shmem = (size_t)(7 * ASZ) * sizeof(float);  // 116,480 B LDS

    E2EGuidedFilter_gfx1250_kernel<<<grid, dim3(256), shmem, stream>>>(x, y, o);
}